// GraphConv_13692355739963
// MI455X (gfx1250) — compile-verified
//
#include <hip/hip_runtime.h>
#include <hip/hip_bf16.h>
#include <cstddef>

#define N_NODES 50000
#define N_EDGES 600000
#define CIN     128
#define COUT    128
#define BATCH   2
#define BO      (BATCH * COUT)     // 256
#define LEAKY   0.3f
#define EDGES_PER_BLOCK 8

typedef float v2f __attribute__((ext_vector_type(2)));
typedef float v8f __attribute__((ext_vector_type(8)));

// ---------------------------------------------------------------------------
// Kernel 1: dual GEMM  F_v = W0*X (stored (B,O,N) -> d_out),
//                      F_n = W1*X (stored TRANSPOSED (N, B*O) -> ws)
// via V_WMMA_F32_16X16X4_F32. Block = 8 waves; block handles one
// (16-node, batch) column; wave w owns output rows [16w, 16w+16).
// X tile (128x16 f32, 8 KB) staged in LDS, B fragment shared by both GEMMs.
// D layout: vgpr r -> row rbase+r, which in the transposed F_n layout is a
// CONTIGUOUS address run -> each lane stores its v8f as 2x b128.
// ---------------------------------------------------------------------------
__global__ __launch_bounds__(256)
void gc_gemm_wmma(const float* __restrict__ X,
                  const float* __restrict__ W0,
                  const float* __restrict__ W1,
                  float* __restrict__ Fv,    // (B, O, N)
                  float* __restrict__ FnT)   // (N, B*O)
{
    __shared__ float ldsX[CIN * 16];          // [i][n], 8 KB

    const int ntile = blockIdx.x;             // 0..3124 (50000/16 exact)
    const int b     = blockIdx.y;             // 0..1
    const int n0    = ntile * 16;
    const int tid   = threadIdx.x;

    // stage X[b][0:128][n0:n0+16] into LDS
    const float* Xb = X + (size_t)b * CIN * N_NODES;
    #pragma unroll
    for (int idx = tid; idx < CIN * 16; idx += 256) {
        const int i = idx >> 4;
        const int n = idx & 15;
        ldsX[idx] = Xb[(size_t)i * N_NODES + n0 + n];
    }
    __syncthreads();

    const int wave = tid >> 5;                // 0..7 -> M tile
    const int lane = tid & 31;
    const int m0   = wave * 16;
    // A (16x4 f32): lanes 0-15 hold M=0..15; vgpr0={K=0|K=2}, vgpr1={K=1|K=3}
    const int arow = m0 + (lane & 15);
    const int lk   = (lane >> 4) << 1;        // 0 or 2
    const int col  = lane & 15;               // B/D column

    v8f c0 = {};                              // W0 accumulator (F_v)
    v8f c1 = {};                              // W1 accumulator (F_n)

    #pragma unroll 4
    for (int kk = 0; kk < CIN; kk += 4) {
        const int kb = kk + lk;
        v2f a0 = *(const v2f*)(W0 + (size_t)arow * CIN + kb);
        v2f a1 = *(const v2f*)(W1 + (size_t)arow * CIN + kb);
        v2f bb;
        bb.x = ldsX[(kb + 0) * 16 + col];
        bb.y = ldsX[(kb + 1) * 16 + col];
        c0 = __builtin_amdgcn_wmma_f32_16x16x4_f32(false, a0, false, bb,
                                                   (short)0, c0, false, false);
        c1 = __builtin_amdgcn_wmma_f32_16x16x4_f32(false, a1, false, bb,
                                                   (short)0, c1, false, false);
    }

    // D layout: vgpr r, lanes 0-15 -> row m0+r, lanes 16-31 -> row m0+8+r
    const int rbase = m0 + ((lane >> 4) << 3);

    // F_v -> (B, O, N): per-r stores coalesced across the 16 lanes of a row
    float* FvB = Fv + (size_t)b * COUT * N_NODES;
    #pragma unroll
    for (int r = 0; r < 8; ++r) {
        FvB[(size_t)(rbase + r) * N_NODES + n0 + col] = c0[r];
    }

    // F_n -> (N, B*O): rows rbase..rbase+7 are contiguous floats; rbase is a
    // multiple of 8 -> 32B-aligned v8f store (lowers to 2x global_store_b128)
    *(v8f*)(FnT + (size_t)(n0 + col) * BO + b * COUT + rbase) = c1;
}

// ---------------------------------------------------------------------------
// Kernel 2: zero segment-sum + count workspace (b128 stores).
// ---------------------------------------------------------------------------
__global__ __launch_bounds__(256)
void gc_zero4(float4* __restrict__ p, int n4)
{
    const int i = blockIdx.x * blockDim.x + threadIdx.x;
    if (i < n4) p[i] = make_float4(0.f, 0.f, 0.f, 0.f);
}

// ---------------------------------------------------------------------------
// Kernel 3: edge scatter. 256-thread block handles EDGES_PER_BLOCK edges.
// Thread t -> channel (b*128+o). With F_n transposed (N, B*O):
//   gather  = Fn[g*256 + t]      (contiguous 1 KB per edge, fully coalesced)
//   scatter = segsum[r*256 + t]  (contiguous 1 KB of atomics per edge)
// ---------------------------------------------------------------------------
__global__ __launch_bounds__(256)
void gc_scatter(const float* __restrict__ FnT,     // (N, B*O)
                const int* __restrict__ gather_index,
                const int* __restrict__ reduce_index,
                float* __restrict__ segsum,        // (N, B*O)
                float* __restrict__ cnt)           // (N)
{
    const int t  = threadIdx.x;                    // 0..255
    const int e0 = blockIdx.x * EDGES_PER_BLOCK;
    #pragma unroll
    for (int k = 0; k < EDGES_PER_BLOCK; ++k) {
        const int e = e0 + k;
        if (e >= N_EDGES) return;
        const int g = gather_index[e];
        const int r = reduce_index[e];
        const float v = FnT[(size_t)g * BO + t];
        atomicAdd(&segsum[(size_t)r * BO + t], v);
        if (t == 0) atomicAdd(&cnt[r], 1.0f);
    }
}

// ---------------------------------------------------------------------------
// Kernel 4: out = leaky(F_v + segsum/max(cnt,1)). d_out already holds F_v.
// Output-coalesced; segsum reads are strided but L2-resident (51 MB << 192 MB).
// ---------------------------------------------------------------------------
__global__ __launch_bounds__(256)
void gc_finalize(float* __restrict__ out,          // (B, O, N)
                 const float* __restrict__ segsum, // (N, B*O)
                 const float* __restrict__ cnt)
{
    const int idx = blockIdx.x * blockDim.x + threadIdx.x;
    if (idx >= BO * N_NODES) return;
    const int n  = idx % N_NODES;
    const int bo = idx / N_NODES;
    const float s = segsum[(size_t)n * BO + bo];
    const float c = cnt[n];
    const float m = s / fmaxf(c, 1.0f);
    const float v = out[idx] + m;
    out[idx] = v > 0.0f ? v : LEAKY * v;
}

extern "C" void kernel_launch(void* const* d_in, const int* in_sizes, int n_in,
                              void* d_out, int out_size, void* d_ws, size_t ws_size,
                              hipStream_t stream)
{
    const float* X  = (const float*)d_in[0];   // (B, C_IN, N)
    const float* W0 = (const float*)d_in[1];   // (O, I)
    const float* W1 = (const float*)d_in[2];   // (O, I)
    const int* ri   = (const int*)d_in[3];     // reduce_index (E)
    const int* gi   = (const int*)d_in[4];     // gather_index (E)
    float* out = (float*)d_out;

    float* ws        = (float*)d_ws;
    const size_t FSZ = (size_t)BO * N_NODES;   // 12.8M floats
    float* FnT       = ws;                     // 51.2 MB, (N, B*O)
    float* segsum    = ws + FSZ;               // 51.2 MB, (N, B*O)
    float* cnt       = segsum + FSZ;           // 0.2 MB  (contiguous after segsum)

    // 1) dual GEMM: F_v -> d_out (B,O,N), F_n -> ws transposed (N,B*O)
    dim3 gridA(N_NODES / 16, BATCH);
    gc_gemm_wmma<<<gridA, 256, 0, stream>>>(X, W0, W1, out, FnT);

    // 2) zero segsum + cnt (contiguous; FSZ and N_NODES are multiples of 4)
    const int zn4 = (int)((FSZ + N_NODES) / 4);
    gc_zero4<<<(zn4 + 255) / 256, 256, 0, stream>>>((float4*)segsum, zn4);

    // 3) edge gather + scatter-add
    const int nblk = (N_EDGES + EDGES_PER_BLOCK - 1) / EDGES_PER_BLOCK;
    gc_scatter<<<nblk, 256, 0, stream>>>(FnT, gi, ri, segsum, cnt);

    // 4) mean + add + LeakyReLU
    const int tot = BO * N_NODES;
    gc_finalize<<<(tot + 255) / 256, 256, 0, stream>>>(out, segsum, cnt);
}